// GNNConv_46943992545896
// MI455X (gfx1250) — compile-verified
//
#include <hip/hip_runtime.h>
#include <stdint.h>

typedef __attribute__((ext_vector_type(2))) float v2f;
typedef __attribute__((ext_vector_type(8))) float v8f;

#define N_NODES   50000
#define N_EDGES   800000
#define D         128
#define N_CLASSES 10
#define KEEP_SCALE 1.11111111f   // 1 / (1 - 0.1)

// ---------------------------------------------------------------------------
// Edge scatter-add: aggr[dst] += x[src], 128 channels per edge.
// Memory-bound random 512B gather + f32 atomic scatter -> global_atomic_add_f32.
// ---------------------------------------------------------------------------
__global__ __launch_bounds__(256)
void scatter_add(const float* __restrict__ x,
                 const int* __restrict__ src,
                 const int* __restrict__ dst,
                 float* __restrict__ aggr) {
    const int e = blockIdx.x * 2 + (threadIdx.x >> 7);   // 2 edges per 256-thread block
    const int c = threadIdx.x & 127;
    if (e < N_EDGES) {
        const int s = src[e];
        const int d = dst[e];
        atomicAdd(&aggr[(size_t)d * D + c], x[(size_t)s * D + c]);
    }
}

// ---------------------------------------------------------------------------
// Layer 1: h = dropout(aggr @ Wrel + x @ Wroot + b), [N,128] -> [N,128]
// One block = 16 rows x 128 cols. 8 waves, one 16x16 tile each.
// f32 WMMA 16x16x4, K=128 -> 32 WMMA pairs per tile.
// ---------------------------------------------------------------------------
__global__ __launch_bounds__(256)
void gemm1_fused(const float* __restrict__ aggr,
                 const float* __restrict__ x,
                 const float* __restrict__ Wrel,
                 const float* __restrict__ Wroot,
                 const float* __restrict__ bias,
                 float* __restrict__ h) {
    const int lane  = threadIdx.x & 31;
    const int wave  = threadIdx.x >> 5;       // 0..7 -> column tile
    const int r0    = blockIdx.x * 16;
    const int n0    = wave * 16;
    const int m     = lane & 15;              // A row / B,C column within tile
    const int khalf = (lane >> 4) * 2;        // upper half-wave carries K+2,K+3
    const int half  = lane >> 4;

    const float* arow = aggr + (size_t)(r0 + m) * D;
    const float* xrow = x    + (size_t)(r0 + m) * D;

    v8f c = {};
    for (int k = 0; k < D; k += 4) {
        const int kk = k + khalf;
        // A fragments (16x4 f32): v2f = {K=kk, K=kk+1} for row m
        v2f aA = *(const v2f*)(arow + kk);
        v2f aX = *(const v2f*)(xrow + kk);
        // B fragments (4x16 f32): v2f = {K=kk, K=kk+1} for column n0+m
        v2f bR, bT;
        bR.x = Wrel [(size_t)kk       * D + n0 + m];
        bR.y = Wrel [(size_t)(kk + 1) * D + n0 + m];
        bT.x = Wroot[(size_t)kk       * D + n0 + m];
        bT.y = Wroot[(size_t)(kk + 1) * D + n0 + m];
        c = __builtin_amdgcn_wmma_f32_16x16x4_f32(false, aA, false, bR,
                                                  (short)0, c, false, false);
        c = __builtin_amdgcn_wmma_f32_16x16x4_f32(false, aX, false, bT,
                                                  (short)0, c, false, false);
    }

    const float bv = bias[n0 + m];
    const int col = n0 + m;
#pragma unroll
    for (int i = 0; i < 8; ++i) {
        const int row = r0 + i + 8 * half;    // C/D layout: VGPR i -> row i (+8 upper half)
        float v = c[i] + bv;
        // deterministic hash-based dropout (p_drop = 0.1)
        uint32_t idx = (uint32_t)(row * D + col);
        uint32_t hs = idx * 2654435761u;
        hs ^= hs >> 16; hs *= 2246822519u; hs ^= hs >> 13;
        v = (hs % 10u != 0u) ? v * KEEP_SCALE : 0.0f;
        h[(size_t)row * D + col] = v;
    }
}

// ---------------------------------------------------------------------------
// Layer 2: out = aggr2 @ Wrel2 + h @ Wroot2 + b2, [N,128] -> [N,10]
// One wave per 16-row tile; 16-col tile zero-padded past class 9.
// ---------------------------------------------------------------------------
__global__ __launch_bounds__(256)
void gemm2_fused(const float* __restrict__ aggr,
                 const float* __restrict__ h,
                 const float* __restrict__ Wrel,
                 const float* __restrict__ Wroot,
                 const float* __restrict__ bias,
                 float* __restrict__ out) {
    const int lane  = threadIdx.x & 31;
    const int wave  = threadIdx.x >> 5;
    const int r0    = (blockIdx.x * 8 + wave) * 16;
    if (r0 >= N_NODES) return;                // uniform per wave
    const int m     = lane & 15;
    const int khalf = (lane >> 4) * 2;
    const int half  = lane >> 4;
    const bool nvalid = (m < N_CLASSES);

    const float* arow = aggr + (size_t)(r0 + m) * D;
    const float* xrow = h    + (size_t)(r0 + m) * D;

    v8f c = {};
    for (int k = 0; k < D; k += 4) {
        const int kk = k + khalf;
        v2f aA = *(const v2f*)(arow + kk);
        v2f aX = *(const v2f*)(xrow + kk);
        v2f bR = {0.0f, 0.0f};
        v2f bT = {0.0f, 0.0f};
        if (nvalid) {                         // divergent loads only; WMMA stays full-EXEC
            bR.x = Wrel [(size_t)kk       * N_CLASSES + m];
            bR.y = Wrel [(size_t)(kk + 1) * N_CLASSES + m];
            bT.x = Wroot[(size_t)kk       * N_CLASSES + m];
            bT.y = Wroot[(size_t)(kk + 1) * N_CLASSES + m];
        }
        c = __builtin_amdgcn_wmma_f32_16x16x4_f32(false, aA, false, bR,
                                                  (short)0, c, false, false);
        c = __builtin_amdgcn_wmma_f32_16x16x4_f32(false, aX, false, bT,
                                                  (short)0, c, false, false);
    }

    if (nvalid) {
        const float bv = bias[m];
#pragma unroll
        for (int i = 0; i < 8; ++i) {
            const int row = r0 + i + 8 * half;
            out[(size_t)row * N_CLASSES + m] = c[i] + bv;
        }
    }
}

// ---------------------------------------------------------------------------
extern "C" void kernel_launch(void* const* d_in, const int* in_sizes, int n_in,
                              void* d_out, int out_size, void* d_ws, size_t ws_size,
                              hipStream_t stream) {
    const float* features = (const float*)d_in[0];
    const int*   edge     = (const int*)  d_in[1];   // [2, N_EDGES]
    const float* Wrel1    = (const float*)d_in[2];
    const float* Wroot1   = (const float*)d_in[3];
    const float* b1       = (const float*)d_in[4];
    const float* Wrel2    = (const float*)d_in[5];
    const float* Wroot2   = (const float*)d_in[6];
    const float* b2       = (const float*)d_in[7];
    float*       out      = (float*)d_out;

    const int* src = edge;
    const int* dst = edge + N_EDGES;

    float* aggr = (float*)d_ws;                          // [N, 128]
    float* h1   = aggr + (size_t)N_NODES * D;            // [N, 128]
    const size_t bytesND = (size_t)N_NODES * D * sizeof(float);

    // Layer 1
    hipMemsetAsync(aggr, 0, bytesND, stream);
    scatter_add<<<N_EDGES / 2, 256, 0, stream>>>(features, src, dst, aggr);
    gemm1_fused<<<N_NODES / 16, 256, 0, stream>>>(aggr, features, Wrel1, Wroot1, b1, h1);

    // Layer 2 (reuse aggr buffer)
    hipMemsetAsync(aggr, 0, bytesND, stream);
    scatter_add<<<N_EDGES / 2, 256, 0, stream>>>(h1, src, dst, aggr);
    gemm2_fused<<<(N_NODES / 16 + 7) / 8, 256, 0, stream>>>(aggr, h1, Wrel2, Wroot2, b2, out);
}